// NonImagingRod_86766929314129
// MI455X (gfx1250) — compile-verified
//
#include <hip/hip_runtime.h>

typedef __attribute__((ext_vector_type(2))) float v2f;
typedef __attribute__((ext_vector_type(8))) float v8f;

#define DAMPING   0.5f
#define MAX_DELTA 1000.0f
#define N_STEPS   31      // 30 no-grad LM iterations + 1 final step

__global__ void rod_init_out(const float* __restrict__ loss_in, float* __restrict__ out) {
    out[0] = loss_in[0];
}

__global__ __launch_bounds__(256) void rod_lm_kernel(
    const float* __restrict__ P, const float* __restrict__ V,
    const float* __restrict__ R, const float* __restrict__ T,
    const float* __restrict__ cbuf, float* __restrict__ out, int n)
{
    const int tid  = blockIdx.x * blockDim.x + threadIdx.x;
    const int lane = threadIdx.x & 31;
    const int m    = lane & 15;       // column / row-within-half index
    const int h    = lane >> 4;       // lane half (K-pair select)
    const float c  = cbuf[0];

    const bool active = (tid < n);
    float px = 0.f, py = 0.f, pz = 0.f, vx = 0.f, vy = 0.f, vz = 0.f;
    if (active) {
        const float* p = P + 3 * tid;
        const float* v = V + 3 * tid;
        px = p[0] - T[0]; py = p[1] - T[1]; pz = p[2] - T[2];
        vx = v[0];        vy = v[1];        vz = v[2];
    }

    // ------------------------------------------------------------------
    // Rotation: P_local = (P-T) @ R, V_local = V @ R via
    // V_WMMA_F32_16X16X4_F32 with A = R^T (16x4 padded), B = rays (4x16).
    // D[m][n] = sum_k R[k][m] * ray_n[k] = local component m of ray n,
    // so lanes 0-15 read x/y/z straight out of D VGPRs 0..2.
    // ------------------------------------------------------------------
    const int wave_base = tid & ~31;
    if (wave_base + 32 <= n) {                     // wave-uniform: EXEC all-1s
        // A operand: lane l holds A[m][K], K = h*2 + vgpr. A[m][k] = R[k][m].
        v2f a;
        a.x = (m < 3) ? R[(h ? 2 : 0) * 3 + m] : 0.0f;   // K = 0 or 2
        a.y = (m < 3 && !h) ? R[3 + m] : 0.0f;           // K = 1 (K=3 row is zero)

        // B operands: lane l holds B[K][n], n = l&15, K = h*2 + vgpr.
        // tile0 = rays of lanes 0..15, tile1 = rays of lanes 16..31.
        float pz_lo = __shfl(pz, m, 32);
        float px_hi = __shfl(px, lane | 16, 32);
        float py_hi = __shfl(py, lane | 16, 32);
        v2f bp0, bp1;
        bp0.x = h ? pz_lo : px;     bp0.y = h ? 0.0f : py;
        bp1.x = h ? pz    : px_hi;  bp1.y = h ? 0.0f : py_hi;

        float vz_lo = __shfl(vz, m, 32);
        float vx_hi = __shfl(vx, lane | 16, 32);
        float vy_hi = __shfl(vy, lane | 16, 32);
        v2f bv0, bv1;
        bv0.x = h ? vz_lo : vx;     bv0.y = h ? 0.0f : vy;
        bv1.x = h ? vz    : vx_hi;  bv1.y = h ? 0.0f : vy_hi;

        v8f zero = {};
        v8f dp0 = __builtin_amdgcn_wmma_f32_16x16x4_f32(false, a, false, bp0, (short)0, zero, false, false);
        v8f dp1 = __builtin_amdgcn_wmma_f32_16x16x4_f32(false, a, false, bp1, (short)0, zero, false, false);
        v8f dv0 = __builtin_amdgcn_wmma_f32_16x16x4_f32(false, a, false, bv0, (short)0, zero, false, false);
        v8f dv1 = __builtin_amdgcn_wmma_f32_16x16x4_f32(false, a, false, bv1, (short)0, zero, false, false);

        // Lanes 0-15 take tile0 directly; lanes 16-31 fetch tile1 column m.
        float px1 = __shfl(dp1[0], m, 32);
        float py1 = __shfl(dp1[1], m, 32);
        float pz1 = __shfl(dp1[2], m, 32);
        float vx1 = __shfl(dv1[0], m, 32);
        float vy1 = __shfl(dv1[1], m, 32);
        float vz1 = __shfl(dv1[2], m, 32);
        px = h ? px1 : dp0[0];  py = h ? py1 : dp0[1];  pz = h ? pz1 : dp0[2];
        vx = h ? vx1 : dv0[0];  vy = h ? vy1 : dv0[1];  vz = h ? vz1 : dv0[2];
    } else if (active) {
        // ragged-tail scalar rotation: row-vector times R
        float qx = px * R[0] + py * R[3] + pz * R[6];
        float qy = px * R[1] + py * R[4] + pz * R[7];
        float qz = px * R[2] + py * R[5] + pz * R[8];
        float wx = vx * R[0] + vy * R[3] + vz * R[6];
        float wy = vx * R[1] + vy * R[4] + vz * R[7];
        float wz = vx * R[2] + vy * R[5] + vz * R[8];
        px = qx; py = qy; pz = qz; vx = wx; vy = wy; vz = wz;
    }

    // ------------------------------------------------------------------
    // f(t) = F(P+tV) is exactly a quadratic in t; hoist the coefficients.
    //   Aq = -c*(vy^2+vz^2);  Bq = vx - 2c*(py*vy+pz*vz);  Cq = px - c*(py^2+pz^2)
    //   f = (Aq*t + Bq)*t + Cq;  f' = 2*Aq*t + Bq
    // ------------------------------------------------------------------
    float acc = 0.0f;
    if (active) {
        float Aq = -c * fmaf(vy, vy, vz * vz);
        float Bq = fmaf(-2.0f * c, fmaf(py, vy, pz * vz), vx);
        float Cq = fmaf(-c, fmaf(py, py, pz * pz), px);
        float A2 = Aq + Aq;
        float t = 0.0f;
        for (int i = 0; i < N_STEPS; ++i) {
            float fp    = fmaf(A2, t, Bq);
            float f     = fmaf(fmaf(Aq, t, Bq), t, Cq);
            float den   = fmaf(fp, fp, DAMPING);       // >= DAMPING > 0
            float delta = f * fp * __builtin_amdgcn_rcpf(den);
            delta = fminf(fmaxf(delta, -MAX_DELTA), MAX_DELTA);
            t -= delta;
        }
        float F = fmaf(fmaf(Aq, t, Bq), t, Cq);        // scale == 1.0
        acc = F * F;
    }

    // wave32 tree reduction, then cross-wave via LDS, then one atomic/block
    for (int off = 16; off > 0; off >>= 1)
        acc += __shfl_down(acc, off, 32);

    __shared__ float smem[8];
    const int warp = threadIdx.x >> 5;
    if (lane == 0) smem[warp] = acc;
    __syncthreads();
    if (threadIdx.x == 0) {
        float s = 0.0f;
        const int nw = (blockDim.x + 31) >> 5;
        for (int i = 0; i < nw; ++i) s += smem[i];
        atomicAdd(out, s * (1.0f / (float)n));
    }
}

extern "C" void kernel_launch(void* const* d_in, const int* in_sizes, int n_in,
                              void* d_out, int out_size, void* d_ws, size_t ws_size,
                              hipStream_t stream) {
    const float* P       = (const float*)d_in[0];
    const float* V       = (const float*)d_in[1];
    const float* R       = (const float*)d_in[2];
    const float* T       = (const float*)d_in[3];
    const float* c       = (const float*)d_in[4];
    const float* loss_in = (const float*)d_in[5];
    float* out = (float*)d_out;

    const int n = in_sizes[0] / 3;

    rod_init_out<<<1, 1, 0, stream>>>(loss_in, out);

    const int block = 256;
    const int grid  = (n + block - 1) / block;
    rod_lm_kernel<<<grid, block, 0, stream>>>(P, V, R, T, c, out, n);
}